// ReluTransformer_84258668412936
// MI455X (gfx1250) — compile-verified
//
#include <hip/hip_runtime.h>

// Problem sizes (fixed by the reference).
#define RROWS 8192
#define NCOLS 4096
#define NP1   4097

// Tiling: each wave handles a 16-row x 256-col stripe.
#define STRIPE    256
#define NSTRIPES  16              // NCOLS / STRIPE
#define ROWTILE   16
#define NROWTILES (RROWS / ROWTILE)   // 512
#define WAVES_PER_BLOCK 8

// Workspace layout (in floats).
#define WS_DLB 0            // diag(w_lb)            [4096]
#define WS_DUB 4096         // diag(w_ub)            [4096]
#define WS_BLB 8192         // bias col of w_lb      [4096]
#define WS_BUB 12288        // bias col of w_ub      [4096]
#define WS_C   16384        // [0]=w_lb[N,N], [1]=w_ub[N,N]
#define WS_P0  16392        // lb row partials       [RROWS * NSTRIPES]
#define WS_P1  (WS_P0 + RROWS * NSTRIPES)

typedef float v2f __attribute__((ext_vector_type(2)));
typedef float v8f __attribute__((ext_vector_type(8)));

// x/out rows are only 4-byte aligned (row stride = 4097 floats).
__device__ __forceinline__ v2f load2u(const float* p) {
    v2f r; __builtin_memcpy(&r, p, 8); return r;
}
__device__ __forceinline__ void store2u(float* p, v2f v) {
    __builtin_memcpy(p, &v, 8);
}

// Gather diag / bias column / corner scalar out of the (structured) w matrices.
__global__ void prep_kernel(const float* __restrict__ wlb,
                            const float* __restrict__ wub,
                            float* __restrict__ ws) {
    int j = blockIdx.x * blockDim.x + threadIdx.x;
    if (j < NCOLS) {
        ws[WS_DLB + j] = wlb[(size_t)j * NP1 + j];
        ws[WS_DUB + j] = wub[(size_t)j * NP1 + j];
        ws[WS_BLB + j] = wlb[(size_t)j * NP1 + NCOLS];
        ws[WS_BUB + j] = wub[(size_t)j * NP1 + NCOLS];
    }
    if (j == 0) {
        ws[WS_C + 0] = wlb[(size_t)NCOLS * NP1 + NCOLS];
        ws[WS_C + 1] = wub[(size_t)NCOLS * NP1 + NCOLS];
    }
}

// out[r,j] = max(x,0)*dA[j] + min(x,0)*dB[j]  for j < N  (elementwise, VALU)
// part[r,stripe] = sum_j max(x,0)*bA[j] + min(x,0)*bB[j] (GEMV, WMMA f32 16x16x4)
//
// All operands are loaded directly in the WMMA 16x4 f32 register layout:
// for a K-chunk starting at column c, lane-half 0 supplies (K0,K1)=cols (c,c+1)
// and lane-half 1 supplies (K2,K3)=cols (c+2,c+3) -> per-lane float2 at
// c + 2*hi. No cross-lane shuffles needed.
__global__ void __launch_bounds__(WAVES_PER_BLOCK * 32)
backsub_stripe_kernel(const float* __restrict__ x,
                      const float* __restrict__ dA, const float* __restrict__ dB,
                      const float* __restrict__ bA, const float* __restrict__ bB,
                      float* __restrict__ out, float* __restrict__ part) {
    const int  lane = threadIdx.x & 31;
    const int  wave = threadIdx.x >> 5;
    const int  task = blockIdx.x * WAVES_PER_BLOCK + wave;
    const int  rowtile = task >> 4;          // / NSTRIPES
    const int  stripe  = task & (NSTRIPES - 1);
    const int  m   = lane & 15;
    const int  hi  = (lane >> 4) & 1;
    const int  baseRow = rowtile * ROWTILE;
    const int  row = baseRow + m;
    const int  lo2 = 2 * hi;                 // lane-half column offset

    const float* xr   = x   + (size_t)row * NP1 + lo2;
    float*       orow = out + (size_t)row * NP1 + lo2;
    const float* dAp  = dA + lo2;
    const float* dBp  = dB + lo2;
    const float* bAp  = bA + lo2;
    const float* bBp  = bB + lo2;

    v8f acc = {0.f, 0.f, 0.f, 0.f, 0.f, 0.f, 0.f, 0.f};

    const int j0 = stripe * STRIPE;
#pragma unroll 4
    for (int it = 0; it < STRIPE / 8; ++it) {
        const int jc = j0 + it * 8;          // 8-col chunk pair per wave-iter

        // This lane's two (K0,K1)/(K2,K3) column pairs.
        v2f x01 = load2u(xr + jc);
        v2f x23 = load2u(xr + jc + 4);
        v2f dA01 = *reinterpret_cast<const v2f*>(dAp + jc);
        v2f dA23 = *reinterpret_cast<const v2f*>(dAp + jc + 4);
        v2f dB01 = *reinterpret_cast<const v2f*>(dBp + jc);
        v2f dB23 = *reinterpret_cast<const v2f*>(dBp + jc + 4);
        v2f b0A  = *reinterpret_cast<const v2f*>(bAp + jc);
        v2f b1A  = *reinterpret_cast<const v2f*>(bAp + jc + 4);
        v2f b0B  = *reinterpret_cast<const v2f*>(bBp + jc);
        v2f b1B  = *reinterpret_cast<const v2f*>(bBp + jc + 4);

        v2f p01, p23, n01, n23, o01, o23;
        p01.x = fmaxf(x01.x, 0.f); p01.y = fmaxf(x01.y, 0.f);
        p23.x = fmaxf(x23.x, 0.f); p23.y = fmaxf(x23.y, 0.f);
        n01.x = fminf(x01.x, 0.f); n01.y = fminf(x01.y, 0.f);
        n23.x = fminf(x23.x, 0.f); n23.y = fminf(x23.y, 0.f);

        o01.x = p01.x * dA01.x + n01.x * dB01.x;
        o01.y = p01.y * dA01.y + n01.y * dB01.y;
        o23.x = p23.x * dA23.x + n23.x * dB23.x;
        o23.y = p23.y * dA23.y + n23.y * dB23.y;
        store2u(orow + jc, o01);
        store2u(orow + jc + 4, o23);

        // Bias GEMV on the matrix pipe; bias replicated across all 16
        // B-columns so every accumulator column holds the row-sum.
        acc = __builtin_amdgcn_wmma_f32_16x16x4_f32(false, p01, false, b0A,
                                                    (short)0, acc, false, false);
        acc = __builtin_amdgcn_wmma_f32_16x16x4_f32(false, n01, false, b0B,
                                                    (short)0, acc, false, false);
        acc = __builtin_amdgcn_wmma_f32_16x16x4_f32(false, p23, false, b1A,
                                                    (short)0, acc, false, false);
        acc = __builtin_amdgcn_wmma_f32_16x16x4_f32(false, n23, false, b1B,
                                                    (short)0, acc, false, false);
    }

    // Every N-column of the 16x16 accumulator is identical: lane with n==0 in
    // each half holds rows baseRow+v (half0) / baseRow+8+v (half1) in VGPR v.
    if ((lane & 15) == 0) {
        const int rbase = baseRow + (hi ? 8 : 0);
#pragma unroll
        for (int v = 0; v < 8; ++v)
            part[(size_t)(rbase + v) * NSTRIPES + stripe] = acc[v];
    }
}

// out[r,N] = sum of stripe partials + max(xN,0)*w_self[N,N] + min(xN,0)*w_other[N,N]
__global__ void finish_kernel(const float* __restrict__ lb,
                              const float* __restrict__ ub,
                              const float* __restrict__ ws,
                              float* __restrict__ out) {
    int idx = blockIdx.x * blockDim.x + threadIdx.x;
    if (idx >= 2 * RROWS) return;
    int mat = (idx >= RROWS) ? 1 : 0;
    int r = idx - mat * RROWS;
    const float* part = ws + (mat ? WS_P1 : WS_P0) + (size_t)r * NSTRIPES;
    float s = 0.f;
#pragma unroll
    for (int i = 0; i < NSTRIPES; ++i) s += part[i];
    float xN = (mat ? ub : lb)[(size_t)r * NP1 + NCOLS];
    float cA = mat ? ws[WS_C + 1] : ws[WS_C + 0];
    float cB = mat ? ws[WS_C + 0] : ws[WS_C + 1];
    out[(size_t)mat * RROWS * NP1 + (size_t)r * NP1 + NCOLS] =
        s + fmaxf(xN, 0.f) * cA + fminf(xN, 0.f) * cB;
}

extern "C" void kernel_launch(void* const* d_in, const int* in_sizes, int n_in,
                              void* d_out, int out_size, void* d_ws, size_t ws_size,
                              hipStream_t stream) {
    (void)in_sizes; (void)n_in; (void)out_size; (void)ws_size;
    const float* lb  = (const float*)d_in[0];
    const float* ub  = (const float*)d_in[1];
    const float* wlb = (const float*)d_in[2];
    const float* wub = (const float*)d_in[3];
    float* out = (float*)d_out;
    float* ws  = (float*)d_ws;

    prep_kernel<<<(NCOLS + 255) / 256, 256, 0, stream>>>(wlb, wub, ws);

    const int blocks = (NROWTILES * NSTRIPES) / WAVES_PER_BLOCK;  // 1024
    // new_lb = lb_pos @ w_lb - lb_neg @ w_ub
    backsub_stripe_kernel<<<blocks, WAVES_PER_BLOCK * 32, 0, stream>>>(
        lb, ws + WS_DLB, ws + WS_DUB, ws + WS_BLB, ws + WS_BUB,
        out, ws + WS_P0);
    // new_ub = ub_pos @ w_ub - ub_neg @ w_lb
    backsub_stripe_kernel<<<blocks, WAVES_PER_BLOCK * 32, 0, stream>>>(
        ub, ws + WS_DUB, ws + WS_DLB, ws + WS_BUB, ws + WS_BLB,
        out + (size_t)RROWS * NP1, ws + WS_P1);

    finish_kernel<<<(2 * RROWS + 255) / 256, 256, 0, stream>>>(lb, ub, ws, out);
}